// Instance_MLP_5025111736832
// MI455X (gfx1250) — compile-verified
//
#include <hip/hip_runtime.h>
#include <math.h>

typedef __attribute__((ext_vector_type(2))) float v2f;
typedef __attribute__((ext_vector_type(8))) float v8f;

#define NT 256           // threads per block everywhere
#define EPS_BN 1e-5f

// monotone float -> uint key (larger float => larger key)
__device__ __forceinline__ unsigned sortkey(float x) {
  unsigned u = __float_as_uint(x);
  return (u & 0x80000000u) ? ~u : (u | 0x80000000u);
}

// ---------------------------------------------------------------------------
// Kernel 1: one workgroup per row. Stream teacher row ONCE into 201KB of LDS
// (MI455X: 320KB/WGP) while computing online-softmax normalizers for teacher
// and student. Then 4-pass 8-bit radix select + compaction + bitonic sort of
// the top-100 entirely from LDS. Emits comb[row, 0:100]=teacher softmax topk,
// comb[row, 100:200]=student softmax gathered at the same indices.
// ---------------------------------------------------------------------------
__global__ void topk_row_kernel(const float* __restrict__ T,
                                const float* __restrict__ S,
                                float* __restrict__ comb,
                                int C, int CIN) {
  extern __shared__ float lrow[];               // C floats (dynamic LDS)
  __shared__ float r_tm[NT], r_ts[NT], r_sm[NT], r_ss[NT];
  __shared__ unsigned hist[256];
  __shared__ unsigned sh_pref;
  __shared__ int sh_need;
  __shared__ float vv[128];
  __shared__ int ixs[128];
  __shared__ int cg, ct;
  __shared__ float s_tmax, s_tinv, s_smax, s_sinv;

  const int row = blockIdx.x, tid = threadIdx.x;
  const int K = CIN >> 1;                       // 100
  const float* trow = T + (size_t)row * C;
  const float* srow = S + (size_t)row * C;

  // ---- pass 1: load row to LDS + online max/sum-exp for both matrices ----
  float tm = -INFINITY, ts = 0.f, sm = -INFINITY, ss = 0.f;
  for (int c = tid; c < C; c += NT) {
    float x = trow[c];
    lrow[c] = x;
    if (x > tm) { ts *= __expf(tm - x); tm = x; }
    ts += __expf(x - tm);
    float y = srow[c];
    if (y > sm) { ss *= __expf(sm - y); sm = y; }
    ss += __expf(y - sm);
  }
  r_tm[tid] = tm; r_ts[tid] = ts; r_sm[tid] = sm; r_ss[tid] = ss;
  __syncthreads();
  for (int off = NT >> 1; off > 0; off >>= 1) {
    if (tid < off) {
      float m1 = r_tm[tid], s1 = r_ts[tid], m2 = r_tm[tid + off], s2 = r_ts[tid + off];
      float m = fmaxf(m1, m2);
      r_tm[tid] = m; r_ts[tid] = s1 * __expf(m1 - m) + s2 * __expf(m2 - m);
      m1 = r_sm[tid]; s1 = r_ss[tid]; m2 = r_sm[tid + off]; s2 = r_ss[tid + off];
      m = fmaxf(m1, m2);
      r_sm[tid] = m; r_ss[tid] = s1 * __expf(m1 - m) + s2 * __expf(m2 - m);
    }
    __syncthreads();
  }
  if (tid == 0) {
    s_tmax = r_tm[0]; s_tinv = 1.f / r_ts[0];
    s_smax = r_sm[0]; s_sinv = 1.f / r_ss[0];
    cg = 0; ct = 0;
  }
  if (tid < 128) { vv[tid] = -INFINITY; ixs[tid] = 0x7FFFFFFF; }
  __syncthreads();

  // ---- radix select: threshold key of K-th largest (all reads from LDS) ----
  unsigned prefix = 0;
  int need = K;
  for (int pass = 0; pass < 4; ++pass) {
    const int shift = 24 - 8 * pass;
    const unsigned maskHi = (pass == 0) ? 0u : (0xFFFFFFFFu << (shift + 8));
    hist[tid] = 0;
    __syncthreads();
    for (int c = tid; c < C; c += NT) {
      unsigned key = sortkey(lrow[c]);
      if ((key & maskHi) == prefix) atomicAdd(&hist[(key >> shift) & 255u], 1u);
    }
    __syncthreads();
    if (tid == 0) {
      unsigned cum = 0;
      for (int b = 255; b >= 0; --b) {
        unsigned hb = hist[b];
        if (cum + hb >= (unsigned)need) {
          need -= (int)cum;
          prefix |= ((unsigned)b) << shift;
          break;
        }
        cum += hb;
      }
      sh_pref = prefix; sh_need = need;
    }
    __syncthreads();
    prefix = sh_pref; need = sh_need;
  }

  // ---- compaction: strictly-greater first, then `need` ties ----
  const unsigned thr = prefix;
  const int G = K - need;                       // exact count of key > thr
  for (int c = tid; c < C; c += NT) {
    float x = lrow[c];
    unsigned key = sortkey(x);
    if (key > thr) {
      int p = atomicAdd(&cg, 1);
      if (p < 128) { vv[p] = x; ixs[p] = c; }
    } else if (key == thr) {
      int t = atomicAdd(&ct, 1);
      if (t < need) { vv[G + t] = x; ixs[G + t] = c; }
    }
  }
  __syncthreads();

  // ---- bitonic sort of 128 slots: desc by value, asc by index (jax order) --
  for (int kk = 2; kk <= 128; kk <<= 1) {
    for (int j = kk >> 1; j > 0; j >>= 1) {
      if (tid < 128) {
        int partner = tid ^ j;
        if (partner > tid) {
          bool desc = ((tid & kk) == 0);
          float va = vv[tid], vb = vv[partner];
          int ia = ixs[tid], ib = ixs[partner];
          bool aFirst = (va > vb) || (va == vb && ia < ib);
          if (desc ? !aFirst : aFirst) {
            vv[tid] = vb; vv[partner] = va;
            ixs[tid] = ib; ixs[partner] = ia;
          }
        }
      }
      __syncthreads();
    }
  }

  // ---- emit comb row: teacher softmax topk ++ student softmax gather ----
  if (tid < K) {
    float tval = vv[tid];
    int idx = ixs[tid];
    comb[(size_t)row * CIN + tid] = __expf(tval - s_tmax) * s_tinv;
    float sl = srow[idx];
    comb[(size_t)row * CIN + K + tid] = __expf(sl - s_smax) * s_sinv;
  }
}

// ---------------------------------------------------------------------------
// Kernel 2: h = comb @ w1^T + b1 via V_WMMA_F32_16X16X4_F32 (exact fp32).
// One wave per 16x16 output tile; K=CIN stepped by 4.
// A lane layout (16x4 f32): lanes 0-15 M=0..15, half0 -> K=k,k+1 ; half1 -> K=k+2,k+3
// B lane layout (4x16 f32): lane holds column n0+l, same K split per half.
// C layout: VGPR i, lane l(+16) -> (M = i + 8*half, N = n0 + l).
// ---------------------------------------------------------------------------
__global__ void gemm_wmma_kernel(const float* __restrict__ comb,
                                 const float* __restrict__ w1,
                                 const float* __restrict__ b1,
                                 float* __restrict__ hbuf,
                                 int M, int N, int Kdim) {
  const int lane = threadIdx.x & 31;
  const int wave = (blockIdx.x * blockDim.x + threadIdx.x) >> 5;
  const int ntN = N >> 4;
  const int m0 = (wave / ntN) << 4;
  const int n0 = (wave % ntN) << 4;
  const int hf = lane >> 4;      // which half of the wave
  const int l  = lane & 15;

  const float* Arow = comb + (size_t)(m0 + l) * Kdim;   // A[m][k], m = m0+l
  const float* Brow = w1   + (size_t)(n0 + l) * Kdim;   // B[k][n] = w1[n][k]

  v8f acc = {};
  for (int k = 0; k < Kdim; k += 4) {
    const int kk = k + (hf << 1);
    const float2 av = *(const float2*)(Arow + kk);
    const float2 bv = *(const float2*)(Brow + kk);
    v2f a; a.x = av.x; a.y = av.y;
    v2f b; b.x = bv.x; b.y = bv.y;
    acc = __builtin_amdgcn_wmma_f32_16x16x4_f32(
        /*neg_a=*/false, a, /*neg_b=*/false, b,
        /*c_mod=*/(short)0, acc, /*reuse_a=*/false, /*reuse_b=*/false);
  }

  const float bias = b1[n0 + l];
#pragma unroll
  for (int i = 0; i < 8; ++i) {
    hbuf[(size_t)(m0 + i + (hf << 3)) * N + (n0 + l)] = acc[i] + bias;
  }
}

// ---------------------------------------------------------------------------
// Kernel 3: deterministic per-column batch stats -> BN scale/shift.
// One block per column; fixed-order tree reduction (no float atomics).
// ---------------------------------------------------------------------------
__global__ void bnstats_kernel(const float* __restrict__ hbuf,
                               const float* __restrict__ gamma,
                               const float* __restrict__ beta,
                               float* __restrict__ scale,
                               float* __restrict__ shift,
                               int M, int N) {
  const int j = blockIdx.x, tid = threadIdx.x;
  float s = 0.f, q = 0.f;
  for (int r = tid; r < M; r += NT) {
    float x = hbuf[(size_t)r * N + j];
    s += x; q += x * x;
  }
  __shared__ float hs[NT], hq[NT];
  hs[tid] = s; hq[tid] = q;
  __syncthreads();
  for (int off = NT >> 1; off > 0; off >>= 1) {
    if (tid < off) { hs[tid] += hs[tid + off]; hq[tid] += hq[tid + off]; }
    __syncthreads();
  }
  if (tid == 0) {
    float mean = hs[0] / (float)M;
    float var  = hq[0] / (float)M - mean * mean;   // biased, matches BN
    float sc = gamma[j] * rsqrtf(var + EPS_BN);
    scale[j] = sc;
    shift[j] = beta[j] - mean * sc;
  }
}

// ---------------------------------------------------------------------------
// Kernel 4: per-row BN + ReLU + dot with conv2_w + bias -> out[row].
// Gradient-reversal layer is identity in forward.
// ---------------------------------------------------------------------------
__global__ void out_kernel(const float* __restrict__ hbuf,
                           const float* __restrict__ scale,
                           const float* __restrict__ shift,
                           const float* __restrict__ w2,
                           const float* __restrict__ b2,
                           float* __restrict__ out, int N) {
  const int row = blockIdx.x, tid = threadIdx.x;
  float x = hbuf[(size_t)row * N + tid];
  float r = fmaxf(fmaf(x, scale[tid], shift[tid]), 0.f);
  __shared__ float hs[NT];
  hs[tid] = r * w2[tid];
  __syncthreads();
  for (int off = NT >> 1; off > 0; off >>= 1) {
    if (tid < off) hs[tid] += hs[tid + off];
    __syncthreads();
  }
  if (tid == 0) out[row] = hs[0] + b2[0];
}

// ---------------------------------------------------------------------------
extern "C" void kernel_launch(void* const* d_in, const int* in_sizes, int n_in,
                              void* d_out, int out_size, void* d_ws, size_t ws_size,
                              hipStream_t stream) {
  const float* T     = (const float*)d_in[0];   // teacher_out [B,C]
  const float* S     = (const float*)d_in[1];   // student_out [B,C]
  // d_in[2] = lambda_ : forward value is identity, unused here
  const float* w1    = (const float*)d_in[3];   // [INTER, CIN]
  const float* b1    = (const float*)d_in[4];   // [INTER]
  const float* gamma = (const float*)d_in[5];
  const float* beta  = (const float*)d_in[6];
  const float* w2    = (const float*)d_in[7];   // [1, INTER]
  const float* b2    = (const float*)d_in[8];   // [1]
  float* out = (float*)d_out;

  const int B     = out_size;                   // 2048
  const int C     = in_sizes[0] / B;            // 50257
  const int INTER = in_sizes[4];                // 256
  const int CIN   = in_sizes[3] / INTER;        // 200

  // workspace layout (floats)
  float* comb = (float*)d_ws;                   // [B, CIN]
  float* hbuf = comb + (size_t)B * CIN;         // [B, INTER]
  float* bnsc = hbuf + (size_t)B * INTER;       // [INTER]
  float* bnsh = bnsc + INTER;                   // [INTER]

  // 1) fused softmax-stats + LDS-resident radix-select top-k + gather
  size_t ldsBytes = (size_t)C * sizeof(float);  // ~201 KB of the 320 KB/WGP
  topk_row_kernel<<<B, NT, ldsBytes, stream>>>(T, S, comb, C, CIN);

  // 2) fp32 WMMA GEMM: h = comb @ w1^T + b1
  int waves = (B >> 4) * (INTER >> 4);          // 2048 tiles
  gemm_wmma_kernel<<<waves / (NT / 32), NT, 0, stream>>>(comb, w1, b1, hbuf, B, INTER, CIN);

  // 3) BN batch stats (deterministic) -> scale/shift
  bnstats_kernel<<<INTER, NT, 0, stream>>>(hbuf, gamma, beta, bnsc, bnsh, B, INTER);

  // 4) BN + ReLU + 256->1 dense + bias
  out_kernel<<<B, INTER, 0, stream>>>(hbuf, bnsc, bnsh, w2, b2, out, INTER);
}